// Net_14044543058087
// MI455X (gfx1250) — compile-verified
//
#include <hip/hip_runtime.h>
#include <hip/hip_bf16.h>

typedef float v2f __attribute__((ext_vector_type(2)));
typedef float v8f __attribute__((ext_vector_type(8)));

#define N_NODES 100000
#define IN_DIM  128
#define HID_DIM 256
#define OUT_DIM 40

// ---------------------------------------------------------------------------
// Edge-parallel mean-aggregation accumulator: one wave32 per edge.
// agg[dst, :] += x[src, :]  (float atomics, D/32 floats per lane)
// cnt[dst]    += 1          (lane 0, only on the first layer)
// ---------------------------------------------------------------------------
template <int D>
__global__ void scatter_accum(const float* __restrict__ x,
                              const int* __restrict__ src,
                              const int* __restrict__ dst,
                              float* __restrict__ agg,
                              float* __restrict__ cnt,
                              int n_edges, int count_deg)
{
    const int lane = threadIdx.x & 31;
    const int e = blockIdx.x * (blockDim.x >> 5) + (threadIdx.x >> 5);
    if (e >= n_edges) return;

    const int s = src[e];
    const int d = dst[e];
    const float* xs = x + (size_t)s * D;
    float* ad = agg + (size_t)d * D;

    constexpr int FPL = D / 32;            // floats per lane: 4 (D=128) or 8 (D=256)
#pragma unroll
    for (int i = 0; i < FPL; i += 4) {
        const float4 v = *(const float4*)(xs + lane * FPL + i);
        atomicAdd(ad + lane * FPL + i + 0, v.x);
        atomicAdd(ad + lane * FPL + i + 1, v.y);
        atomicAdd(ad + lane * FPL + i + 2, v.z);
        atomicAdd(ad + lane * FPL + i + 3, v.w);
    }
    if (count_deg && lane == 0) atomicAdd(cnt + d, 1.0f);
}

// ---------------------------------------------------------------------------
// Fused SAGEConv GEMM: out[16 x 16*NT strip] = (agg/deg) @ Wl + x @ Wr + b
// One wave32 per strip; K-loop of V_WMMA_F32_16X16X4_F32 with the A fragment
// reused across NT column tiles (NT accumulators).
//
// A fragment (16x4 f32, ISA layout): lane L holds row M = L&15;
//   VGPR0/1 = K = {0,1} for lanes 0-15, K = {2,3} for lanes 16-31.
// C/D (16x16 f32, 8 VGPRs): lane column N = L&15; VGPR r -> row r (+8 hi half).
// EXEC is all 1s through every WMMA (no divergence before the final store).
// K / NOUT are compile-time so all inner-loop addressing folds to immediates.
// ---------------------------------------------------------------------------
template <int K, int NOUT, int NT, bool RELU>
__global__ void sage_fused_gemm(const float* __restrict__ agg,
                                const float* __restrict__ cnt,
                                const float* __restrict__ xin,
                                const float* __restrict__ Wl,   // [K x NOUT]
                                const float* __restrict__ Wr,   // [K x NOUT]
                                const float* __restrict__ bias, // [NOUT]
                                float* __restrict__ out)        // [n x NOUT]
{
    const int m0   = blockIdx.x * 16;
    const int n0   = blockIdx.y * (16 * NT);
    const int lane = threadIdx.x & 31;
    const int mrow = lane & 15;
    const int kh   = (lane >> 4) * 2;          // K sub-offset within a step: 0 or 2
    const int ncol = lane & 15;

    const int m = m0 + mrow;
    const float inv = 1.0f / fmaxf(cnt[m], 1.0f);

    int nc[NT];
#pragma unroll
    for (int t = 0; t < NT; ++t)
        nc[t] = min(n0 + 16 * t + ncol, NOUT - 1);   // clamped; store masked later

    const float* arow = agg + (size_t)m * K;
    const float* xrow = xin + (size_t)m * K;

    v8f acc[NT];
#pragma unroll
    for (int t = 0; t < NT; ++t) acc[t] = (v8f){};

    // pass 1: (agg * 1/deg) @ Wl  — per-row scale folded into the A fragment
#pragma unroll 8
    for (int kk = 0; kk < K; kk += 4) {
        v2f a;
        a.x = arow[kk + kh + 0] * inv;
        a.y = arow[kk + kh + 1] * inv;
#pragma unroll
        for (int t = 0; t < NT; ++t) {
            v2f b;
            b.x = Wl[(kk + kh + 0) * NOUT + nc[t]];
            b.y = Wl[(kk + kh + 1) * NOUT + nc[t]];
            acc[t] = __builtin_amdgcn_wmma_f32_16x16x4_f32(false, a, false, b,
                                                           (short)0, acc[t],
                                                           false, false);
        }
    }
    // pass 2: x @ Wr
#pragma unroll 8
    for (int kk = 0; kk < K; kk += 4) {
        v2f a;
        a.x = xrow[kk + kh + 0];
        a.y = xrow[kk + kh + 1];
#pragma unroll
        for (int t = 0; t < NT; ++t) {
            v2f b;
            b.x = Wr[(kk + kh + 0) * NOUT + nc[t]];
            b.y = Wr[(kk + kh + 1) * NOUT + nc[t]];
            acc[t] = __builtin_amdgcn_wmma_f32_16x16x4_f32(false, a, false, b,
                                                           (short)0, acc[t],
                                                           false, false);
        }
    }

#pragma unroll
    for (int t = 0; t < NT; ++t) {
        const float bv  = bias[nc[t]];
        const int   col = n0 + 16 * t + ncol;
        if (col < NOUT) {
#pragma unroll
            for (int r = 0; r < 8; ++r) {
                const int row = m0 + r + 4 * kh;   // +8 for the hi half-wave
                float v = acc[t][r] + bv;
                if (RELU) v = fmaxf(v, 0.0f);
                out[(size_t)row * NOUT + col] = v;
            }
        }
    }
}

// ---------------------------------------------------------------------------
extern "C" void kernel_launch(void* const* d_in, const int* in_sizes, int n_in,
                              void* d_out, int out_size, void* d_ws, size_t ws_size,
                              hipStream_t stream)
{
    const float* x   = (const float*)d_in[0];
    const int*   ei  = (const int*)  d_in[1];   // [2, E] -> [src | dst]
    const float* W1l = (const float*)d_in[2];
    const float* b1  = (const float*)d_in[3];
    const float* W1r = (const float*)d_in[4];
    const float* W2l = (const float*)d_in[5];
    const float* b2  = (const float*)d_in[6];
    const float* W2r = (const float*)d_in[7];
    float* out = (float*)d_out;

    const int E = in_sizes[1] / 2;
    const int* src = ei;
    const int* dst = ei + E;

    // workspace layout (floats): [agg1 N*128][cnt N][agg2 N*256][h N*256]
    float* ws   = (float*)d_ws;
    float* agg1 = ws;
    float* cnt  = agg1 + (size_t)N_NODES * IN_DIM;
    float* agg2 = cnt + N_NODES;
    float* h    = agg2 + (size_t)N_NODES * HID_DIM;

    // zero agg1 + cnt + agg2 (h is fully overwritten by the layer-1 GEMM)
    const size_t zero_floats =
        (size_t)N_NODES * IN_DIM + N_NODES + (size_t)N_NODES * HID_DIM;
    hipMemsetAsync(d_ws, 0, zero_floats * sizeof(float), stream);

    const int WPB = 8;                        // waves (edges) per 256-thread block
    const dim3 sblk(32 * WPB);
    const dim3 sgrd((E + WPB - 1) / WPB);

    // ---- layer 1: agg -> h = relu(mean@W1l + b1 + x@W1r) ----
    scatter_accum<IN_DIM><<<sgrd, sblk, 0, stream>>>(x, src, dst, agg1, cnt, E, 1);
    {
        dim3 grd(N_NODES / 16, HID_DIM / (16 * 4));   // 6250 x 4
        sage_fused_gemm<IN_DIM, HID_DIM, 4, true>
            <<<grd, dim3(32), 0, stream>>>(agg1, cnt, x, W1l, W1r, b1, h);
    }

    // ---- layer 2: agg -> out = mean@W2l + b2 + h@W2r ----
    scatter_accum<HID_DIM><<<sgrd, sblk, 0, stream>>>(h, src, dst, agg2, cnt, E, 0);
    {
        dim3 grd(N_NODES / 16, 1);                    // 3 tiles cover 48 >= 40 cols
        sage_fused_gemm<HID_DIM, OUT_DIM, 3, false>
            <<<grd, dim3(32), 0, stream>>>(agg2, cnt, h, W2l, W2r, b2, out);
    }
}